// OnlineTripletLoss_62191126446259
// MI455X (gfx1250) — compile-verified
//
#include <hip/hip_runtime.h>
#include <hip/hip_bf16.h>

// OnlineTripletLoss for MI455X (gfx1250, wave32).
//
// Strategy: gather-bound problem (384 MB of L2-resident gathers, 64 MB table
// fits in the 192 MB L2; ~134 MFLOP of dot products). Keep everything fp32 and
// use V_WMMA_F32_16X16X4_F32 so the dot-product MACs run on the matrix pipe
// while VMEM streams gathers. One wave handles 16 triplets: accumulate the
// 16x16 fp32 tiles A*P^T and A*N^T over K=256 in chunks of 4, read only the
// diagonals. The 16x16x4 f32 A-operand layout (lane L: row L%16, K pair
// 2*(L/16)) means each operand chunk is a single 8-byte contiguous load per
// lane from the gathered row -- no transpose, no LDS staging.

typedef __attribute__((ext_vector_type(2))) float v2f;
typedef __attribute__((ext_vector_type(8))) float v8f;

#define EMB_D 256
#define INV_MARGIN (1.0f / 0.7f)

__global__ __launch_bounds__(256)
void triplet_wmma_kernel(const float* __restrict__ emb,
                         const int* __restrict__ trip,
                         float* __restrict__ ws,   // [gridDim.x * 3] partials
                         int T)
{
    const int L         = threadIdx.x & 31;     // lane in wave32
    const int waveInBlk = threadIdx.x >> 5;     // 0..7
    const int wavesPerBlk = blockDim.x >> 5;
    const int wave      = blockIdx.x * wavesPerBlk + waveInBlk;
    const int nwaves    = gridDim.x * wavesPerBlk;
    const int ngroups   = T >> 4;               // 16 triplets per group
    const int r         = L & 15;               // triplet row within group
    const int colBase   = (L >> 4) << 1;        // lanes 16-31 start at K=2

    float sexp = 0.f, sap = 0.f, san = 0.f;

    for (int g = wave; g < ngroups; g += nwaves) {
        const int t  = (g << 4) + r;
        const int ia = trip[3 * t + 0];
        const int ip = trip[3 * t + 1];
        const int in_ = trip[3 * t + 2];
        const float* pa = emb + (size_t)ia * EMB_D + colBase;
        const float* pp = emb + (size_t)ip * EMB_D + colBase;
        const float* pn = emb + (size_t)in_ * EMB_D + colBase;

        v8f accp = {};   // 16x16 fp32 tile: A * P^T
        v8f accn = {};   // 16x16 fp32 tile: A * N^T
#pragma unroll 8
        for (int c = 0; c < EMB_D / 4; ++c) {
            // Per K-chunk of 4: lane L supplies (row L%16, K = 4c+2*(L/16)+{0,1})
            // for the A operand; the B operand (P^T / N^T tile) has the mirrored
            // lane layout, so the identical load pattern feeds it too.
            v2f av = *(const v2f*)(pa + 4 * c);
            v2f pv = *(const v2f*)(pp + 4 * c);
            v2f nv = *(const v2f*)(pn + 4 * c);
            accp = __builtin_amdgcn_wmma_f32_16x16x4_f32(
                false, av, false, pv, (short)0, accp, false, false);
            accn = __builtin_amdgcn_wmma_f32_16x16x4_f32(
                false, av, false, nv, (short)0, accn, false, false);
        }

        // Diagonal extraction (C/D layout: VGPR j, lanes 0-15 -> M=j,N=lane;
        // lanes 16-31 -> M=8+j, N=lane-16). Diag t<8 lives at lane t, c[t];
        // diag t>=8 lives at lane t+16, c[t-8]. Branchless so EXEC stays all-1s
        // for the next group's WMMAs.
        float dp = 0.f, dn = 0.f;
#pragma unroll
        for (int j = 0; j < 8; ++j) {
            const bool sel = (L == j) || (L == 24 + j);
            dp = sel ? accp[j] : dp;
            dn = sel ? accn[j] : dn;
        }
        const float m = ((L < 8) || (L >= 24)) ? 1.f : 0.f;
        dp = fminf(fmaxf(dp, -1.f), 1.f);
        dn = fminf(fmaxf(dn, -1.f), 1.f);
        sexp += m * __expf((dn - dp) * INV_MARGIN);
        sap  += m * (1.f - dp);
        san  += m * (1.f - dn);
    }

    // Deterministic wave32 tree reduce.
#pragma unroll
    for (int off = 16; off > 0; off >>= 1) {
        sexp += __shfl_xor(sexp, off, 32);
        sap  += __shfl_xor(sap,  off, 32);
        san  += __shfl_xor(san,  off, 32);
    }

    __shared__ float sw[8][3];
    if (L == 0) {
        sw[waveInBlk][0] = sexp;
        sw[waveInBlk][1] = sap;
        sw[waveInBlk][2] = san;
    }
    __syncthreads();
    if (threadIdx.x == 0) {
        float a = 0.f, b = 0.f, c = 0.f;
        for (int w = 0; w < wavesPerBlk; ++w) {
            a += sw[w][0]; b += sw[w][1]; c += sw[w][2];
        }
        ws[blockIdx.x * 3 + 0] = a;
        ws[blockIdx.x * 3 + 1] = b;
        ws[blockIdx.x * 3 + 2] = c;
    }
}

__global__ __launch_bounds__(256)
void triplet_reduce_kernel(const float* __restrict__ ws,
                           float* __restrict__ out,
                           int nblocks, float invT)
{
    __shared__ float s0[256], s1[256], s2[256];
    const int t = threadIdx.x;
    float a = 0.f, b = 0.f, c = 0.f;
    for (int i = t; i < nblocks; i += 256) {
        a += ws[3 * i + 0];
        b += ws[3 * i + 1];
        c += ws[3 * i + 2];
    }
    s0[t] = a; s1[t] = b; s2[t] = c;
    __syncthreads();
#pragma unroll
    for (int off = 128; off > 0; off >>= 1) {
        if (t < off) {
            s0[t] += s0[t + off];
            s1[t] += s1[t + off];
            s2[t] += s2[t + off];
        }
        __syncthreads();
    }
    if (t == 0) {
        out[0] = s0[0] * invT;   // mean exp((an-ap)/margin)
        out[1] = s1[0] * invT;   // mean (1 - ap_d)
        out[2] = s2[0] * invT;   // mean (1 - an_d)
    }
}

extern "C" void kernel_launch(void* const* d_in, const int* in_sizes, int n_in,
                              void* d_out, int out_size, void* d_ws, size_t ws_size,
                              hipStream_t stream) {
    const float* emb  = (const float*)d_in[0];   // [65536, 256] fp32
    const int*   trip = (const int*)d_in[1];     // [131072, 3]  int32
    float* out = (float*)d_out;                  // 3 fp32 scalars
    float* ws  = (float*)d_ws;                   // block partials (3 fp32 each)

    const int T = in_sizes[1] / 3;
    const int blocks = 256;                      // 2048 waves, 4 groups each
    triplet_wmma_kernel<<<blocks, 256, 0, stream>>>(emb, trip, ws, T);
    triplet_reduce_kernel<<<1, 256, 0, stream>>>(ws, out, blocks,
                                                 1.0f / (float)T);
}